// KNNInterpolationONNX_4612794875948
// MI455X (gfx1250) — compile-verified
//
#include <hip/hip_runtime.h>
#include <math.h>

typedef __attribute__((ext_vector_type(2))) float v2f;
typedef __attribute__((ext_vector_type(8))) float v8f;

#define EPS_KNN 1e-8f
#define BIGF 3.0e38f

// ---------------------------------------------------------------------------
// Prepass: pack sources as float4 { -2x, -2y, -2z, x^2+y^2+z^2 }.
// Padded entries (i >= N) get {0,0,0,+BIG} so they never win the top-3.
// ---------------------------------------------------------------------------
__global__ void knn_prep_sources(const float* __restrict__ sp,
                                 float4* __restrict__ P,
                                 int N, int Npad) {
  int i = blockIdx.x * blockDim.x + threadIdx.x;
  if (i >= Npad) return;
  if (i < N) {
    float x = sp[3 * i + 0];
    float y = sp[3 * i + 1];
    float z = sp[3 * i + 2];
    P[i] = make_float4(-2.f * x, -2.f * y, -2.f * z, x * x + y * y + z * z);
  } else {
    P[i] = make_float4(0.f, 0.f, 0.f, BIGF);
  }
}

// ---------------------------------------------------------------------------
// Main kernel: one wave32 handles 16 targets.
// Per source tile of 16: one V_WMMA_F32_16X16X4_F32 with
//   A = [-2x,-2y,-2z,s2] (sources), B = [tx,ty,tz,1] (targets), C init = t2
// yields the true squared distance d[m,n] = ||s-t||^2 for a 16x16 tile.
// Each lane keeps a top-3 over its 8 rows; a wave-uniform ballot skips the
// insert ladder when no lane's tile minimum beats its 3rd-best.
// ---------------------------------------------------------------------------
__global__ __launch_bounds__(256) void knn_interp_wmma(
    const float4* __restrict__ P,   // prepacked sources [Npad]
    const float* __restrict__ F,    // source features   [N, C]
    const float* __restrict__ T,    // target points     [M, 3]
    float* __restrict__ out,        // output            [M, C]
    int M, int C, int Npad) {
  const int lane = threadIdx.x & 31;
  const int waveId = blockIdx.x * (blockDim.x >> 5) + (threadIdx.x >> 5);
  const int tBase = waveId * 16;          // 16 targets per wave
  const bool hi = lane >= 16;
  const int col = lane & 15;              // target column within tile
  const int n = tBase + col;
  const int nc = (n < M) ? n : (M - 1);   // clamped load index (no divergence)

  // B operand (loop-invariant): lanes 0-15 hold (K0,K1)=(tx,ty),
  // lanes 16-31 hold (K2,K3)=(tz,1.0)
  float tx = T[3 * nc + 0];
  float ty = T[3 * nc + 1];
  float tz = T[3 * nc + 2];
  float t2 = tx * tx + ty * ty + tz * tz;
  v2f b;
  b.x = hi ? tz : tx;
  b.y = hi ? 1.0f : ty;

  // running top-3 true squared distances
  float d0 = BIGF, d1 = BIGF, d2 = BIGF;
  int i0 = 0, i1 = 0, i2 = 0;

  // branchless sorted insert of (d, idx) into (d0<=d1<=d2)
  auto insert3 = [&](float d, int idx) {
    bool c0 = d < d0, c1 = d < d1, c2 = d < d2;
    float nd2 = c1 ? d1 : (c2 ? d : d2);
    int ni2 = c1 ? i1 : (c2 ? idx : i2);
    float nd1 = c0 ? d0 : (c1 ? d : d1);
    int ni1 = c0 ? i0 : (c1 ? idx : i1);
    float nd0 = c0 ? d : d0;
    int ni0 = c0 ? idx : i0;
    d0 = nd0; d1 = nd1; d2 = nd2;
    i0 = ni0; i1 = ni1; i2 = ni2;
  };

  // A operand per tile: lane reads 8 bytes of the float4:
  // lanes 0-15 -> (-2x,-2y), lanes 16-31 -> (-2z, s2)
  const float2* Phalf = reinterpret_cast<const float2*>(P);
  const int numTiles = Npad >> 4;
  const int aOff = 2 * col + (hi ? 1 : 0);  // float2 index within tile*32

  float2 av = Phalf[aOff];  // prefetched tile 0
  for (int t = 0; t < numTiles; ++t) {
    // prefetch next tile's A fragment (clamped, uniform, no divergence)
    const int tnext = (t + 1 < numTiles) ? (t + 1) : t;
    float2 avn = Phalf[tnext * 32 + aOff];

    v2f a;
    a.x = av.x;
    a.y = av.y;
    v8f c = {t2, t2, t2, t2, t2, t2, t2, t2};  // fold ||t||^2 into accumulator
    c = __builtin_amdgcn_wmma_f32_16x16x4_f32(
        /*neg_a=*/false, a, /*neg_b=*/false, b,
        /*c_mod=*/(short)0, c, /*reuse_a=*/false, /*reuse_b=*/false);
    av = avn;

    // wave-uniform skip: can any of this lane's 8 candidates enter its top-3?
    float minv = fminf(fminf(fminf(c[0], c[1]), fminf(c[2], c[3])),
                       fminf(fminf(c[4], c[5]), fminf(c[6], c[7])));
    if (__any((int)(minv < d2))) {
      const int idxBase = t * 16 + (hi ? 8 : 0);
#pragma unroll
      for (int v = 0; v < 8; ++v) {
        insert3(c[v], idxBase + v);
      }
    }
  }

  // merge the two lane halves (rows 0-7 vs rows 8-15 of each tile)
  {
    float pa = __shfl_xor(d0, 16, 32);
    float pb = __shfl_xor(d1, 16, 32);
    float pc = __shfl_xor(d2, 16, 32);
    int ja = __shfl_xor(i0, 16, 32);
    int jb = __shfl_xor(i1, 16, 32);
    int jc = __shfl_xor(i2, 16, 32);
    insert3(pa, ja);
    insert3(pb, jb);
    insert3(pc, jc);
  }

  // inverse-distance weights (d0..d2 are true squared distances already)
  float w0 = 1.0f / (d0 + EPS_KNN);
  float w1 = 1.0f / (d1 + EPS_KNN);
  float w2 = 1.0f / (d2 + EPS_KNN);
  float wsum = w0 + w1 + w2;
  w0 /= wsum; w1 /= wsum; w2 /= wsum;

  // cooperative feature gather+blend: whole wave serves each of 16 targets
  for (int tt = 0; tt < 16; ++tt) {
    int tn = tBase + tt;
    if (tn >= M) break;
    float ww0 = __shfl(w0, tt, 32);
    float ww1 = __shfl(w1, tt, 32);
    float ww2 = __shfl(w2, tt, 32);
    int j0 = __shfl(i0, tt, 32);
    int j1 = __shfl(i1, tt, 32);
    int j2 = __shfl(i2, tt, 32);
    const float* f0 = F + (size_t)j0 * C;
    const float* f1 = F + (size_t)j1 * C;
    const float* f2 = F + (size_t)j2 * C;
    float* o = out + (size_t)tn * C;
    for (int ch = lane; ch < C; ch += 32) {
      o[ch] = ww0 * f0[ch] + ww1 * f1[ch] + ww2 * f2[ch];
    }
  }
}

// ---------------------------------------------------------------------------
// Harness entry
// ---------------------------------------------------------------------------
extern "C" void kernel_launch(void* const* d_in, const int* in_sizes, int n_in,
                              void* d_out, int out_size, void* d_ws,
                              size_t ws_size, hipStream_t stream) {
  const float* sp = (const float*)d_in[0];  // source_points  [N,3]
  const float* sf = (const float*)d_in[1];  // source_features[N,C]
  const float* tp = (const float*)d_in[2];  // target_points  [M,3]
  float* out = (float*)d_out;               // [M,C]

  const int N = in_sizes[0] / 3;
  const int M = in_sizes[2] / 3;
  const int C = in_sizes[1] / N;
  const int Npad = (N + 15) & ~15;

  float4* P = (float4*)d_ws;  // needs Npad*16 bytes (256 KB for N=16384)

  {
    dim3 blk(256), grd((Npad + 255) / 256);
    hipLaunchKernelGGL(knn_prep_sources, grd, blk, 0, stream, sp, P, N, Npad);
  }
  {
    const int waves = (M + 15) / 16;
    const int wavesPerBlock = 8;  // 256 threads
    const int blocks = (waves + wavesPerBlock - 1) / wavesPerBlock;
    hipLaunchKernelGGL(knn_interp_wmma, dim3(blocks), dim3(256), 0, stream, P,
                       sf, tp, out, M, C, Npad);
  }
}